// NearestEmbed_84121229459553
// MI455X (gfx1250) — compile-verified
//
#include <hip/hip_runtime.h>

typedef float v2f __attribute__((ext_vector_type(2)));
typedef float v4f __attribute__((ext_vector_type(4)));
typedef float v8f __attribute__((ext_vector_type(8)));

#define HW       1024      // 32*32 spatial positions
#define K_EMB    512       // codebook entries
#define GDIM     64        // group (vector) dimension
#define PACK_F   32768     // packed codebook floats (64*512, permuted)

// Packed B-fragment layout (floats):
//   fidx(c,k) = ((nt*8 + ktpair)*32 + n + hoff)*4 + sub
//   nt = k>>4, n = k&15, kt = c>>2, ktpair = kt>>1, odd = kt&1,
//   rem = c&3, hoff = (rem&2)<<3, sub = (odd<<1)|(rem&1)
// => per (nt,ktpair) a wave reads 32 lanes * 16B contiguous: one ds_load_b128
//    per lane supplies B fragments for TWO chained WMMAs.

__global__ __launch_bounds__(256)
void nearest_embed_wmma_kernel(const float* __restrict__ x,
                               const float* __restrict__ w,
                               float* __restrict__ out)
{
    extern __shared__ float smem[];
    float* s_pack = smem;             // 32768 floats: codebook in fragment order
    float* s_ee   = smem + PACK_F;    // 512 floats:  ||e_k||^2

    const int t    = threadIdx.x;
    const int lane = t & 31;
    const int wave = t >> 5;

    // ---- stage codebook global -> LDS, permuting into WMMA fragment order ----
    {
        const float4* wsrc = (const float4*)w;           // 8192 float4s
        #pragma unroll
        for (int i = 0; i < 32; ++i) {
            int idx  = i * 256 + t;
            int c    = idx >> 7;                  // row (channel), 128 float4/row
            int col4 = idx & 127;
            float4 v = wsrc[c * 128 + col4];
            int k0     = col4 << 2;
            int nt     = k0 >> 4;
            int nbase  = k0 & 15;
            int kt     = c >> 2;
            int ktpair = kt >> 1;
            int rem    = c & 3;
            int hoff   = (rem & 2) << 3;          // 0 or 16
            int sub    = ((kt & 1) << 1) | (rem & 1);
            int f0 = ((nt * 8 + ktpair) * 32 + nbase + hoff) * 4 + sub;
            s_pack[f0]      = v.x;                // j stride = 4 floats (16B)
            s_pack[f0 + 4]  = v.y;
            s_pack[f0 + 8]  = v.z;
            s_pack[f0 + 12] = v.w;
        }
    }
    // ---- ||e_k||^2 straight from global (L2-hot), 2 columns per thread ----
    #pragma unroll
    for (int i = 0; i < 2; ++i) {
        int k = i * 256 + t;
        float s = 0.f;
        #pragma unroll
        for (int c = 0; c < GDIM; ++c) {
            float e = w[c * K_EMB + k];
            s += e * e;
        }
        s_ee[k] = s;
    }
    __syncthreads();

    // ---- block -> (batch, group, 128-position chunk) ----
    const int blk   = blockIdx.x;
    const int chunk = blk & 7;
    const int grp   = (blk >> 3) & 3;
    const int b     = blk >> 5;
    const int hw0   = chunk * 128 + wave * 16;

    const float* xg = x + (size_t)(b * 256 + grp * 64) * HW;

    // ---- preload A panel: 16 positions x 64 K (fp32 16x4 fragments) ----
    // lane<16 holds K=k0,k0+1 for M=lane ; lane>=16 holds K=k0+2,k0+3
    const int m     = lane & 15;
    const int khalf = (lane >> 4) << 1;           // 0 or 2
    const int pos   = hw0 + m;
    v2f a[16];
    #pragma unroll
    for (int kt = 0; kt < 16; ++kt) {
        int kk  = kt * 4 + khalf;
        a[kt].x = xg[(size_t)kk * HW + pos];
        a[kt].y = xg[(size_t)(kk + 1) * HW + pos];
    }

    float best[8];
    int   bidx[8];
    #pragma unroll
    for (int j = 0; j < 8; ++j) { best[j] = 3.4e38f; bidx[j] = 0; }

    const float* bbase = s_pack + lane * 4;       // + nt*1024 + ktpair*128

    // ---- sweep 32 N-tiles; 8 x ds_load_b128 feed 16 fp32 WMMAs each ----
    #pragma unroll 1
    for (int nt = 0; nt < 32; ++nt) {
        const int col = nt * 16 + (lane & 15);
        const float* bnt = bbase + nt * 1024;
        v8f acc0 = {};
        v8f acc1 = {};
        #pragma unroll
        for (int ktp = 0; ktp < 8; ++ktp) {
            v4f bq = *(const v4f*)(bnt + ktp * 128);      // one b128: 2 fragments
            v2f b0 = __builtin_shufflevector(bq, bq, 0, 1);
            v2f b1 = __builtin_shufflevector(bq, bq, 2, 3);
            acc0 = __builtin_amdgcn_wmma_f32_16x16x4_f32(
                       false, a[2 * ktp],     false, b0, (short)0, acc0, false, false);
            acc1 = __builtin_amdgcn_wmma_f32_16x16x4_f32(
                       false, a[2 * ktp + 1], false, b1, (short)0, acc1, false, false);
        }
        const float ee = s_ee[col];
        #pragma unroll
        for (int j = 0; j < 8; ++j) {
            float score = __builtin_fmaf(-2.0f, acc0[j] + acc1[j], ee);
            if (score < best[j]) { best[j] = score; bidx[j] = col; }
        }
    }

    // ---- argmin across the 16 column-lanes of each half (wave32 butterflies) ----
    #pragma unroll
    for (int j = 0; j < 8; ++j) {
        #pragma unroll
        for (int off = 1; off < 16; off <<= 1) {
            float os = __shfl_xor(best[j], off, 32);
            int   oi = __shfl_xor(bidx[j], off, 32);
            if (os < best[j] || (os == best[j] && oi < bidx[j])) {
                best[j] = os; bidx[j] = oi;       // jnp.argmin: lowest index wins
            }
        }
    }
    int obidx[8];
    #pragma unroll
    for (int j = 0; j < 8; ++j) obidx[j] = __shfl_xor(bidx[j], 16, 32);

    // ---- route k* of row p = lane%16 to this lane ----
    const int p  = lane & 15;
    const int p7 = p & 7;
    int va = bidx[0], vb = obidx[0];
    #pragma unroll
    for (int j = 1; j < 8; ++j) {
        va = (p7 == j) ? bidx[j]  : va;
        vb = (p7 == j) ? obidx[j] : vb;
    }
    const bool own = ((p >> 3) == (lane >> 4));
    const int kst = own ? va : vb;

    // ---- write argmin (as float) ----
    float* argout = out + (size_t)16777216 + (size_t)(b * 4 + grp) * HW + hw0;
    if (lane < 16) argout[lane] = (float)kst;

    // ---- write result: out[b, grp*64+c, pos] = emb[c][k*] * 0.5 ----
    // gather from packed layout via inverse index map (constants fold per c)
    const int gnt = kst >> 4;
    const int gn  = kst & 15;
    const int h   = lane >> 4;
    const float* gbase = s_pack + gnt * 1024 + gn * 4;    // + per-c constant
    float* ro = out + (size_t)(b * 256 + grp * 64) * HW + hw0 + p;
    #pragma unroll
    for (int c2 = 0; c2 < 32; ++c2) {
        int c      = c2 * 2 + h;
        int kt     = c >> 2;
        int ktpair = kt >> 1;
        int rem    = c & 3;
        int hoff   = (rem & 2) << 3;
        int sub    = ((kt & 1) << 1) | (rem & 1);
        int foff   = (ktpair * 32 + hoff) * 4 + sub;      // compile-time per c2
        ro[(size_t)c * HW] = 0.5f * gbase[foff];
    }
}

extern "C" void kernel_launch(void* const* d_in, const int* in_sizes, int n_in,
                              void* d_out, int out_size, void* d_ws, size_t ws_size,
                              hipStream_t stream) {
    const float* x = (const float*)d_in[0];   // (64,256,32,32) fp32
    const float* w = (const float*)d_in[1];   // (64,512) fp32
    float* out = (float*)d_out;               // result (64,256,32,32) ++ argmin (64,4,32,32)

    const size_t shmem = (size_t)(PACK_F + K_EMB) * sizeof(float);  // 130 KB
    nearest_embed_wmma_kernel<<<dim3(2048), dim3(256), shmem, stream>>>(x, w, out);
}